// TypedMessagePassingLayer_19292993093994
// MI455X (gfx1250) — compile-verified
//
#include <hip/hip_runtime.h>
#include <cstdint>
#include <cstddef>

#define D 128
#define LDSP 132          // padded LDS row stride (floats) to dodge bank conflicts
#define NREL 8

typedef float v2f __attribute__((ext_vector_type(2)));
typedef float v8f __attribute__((ext_vector_type(8)));

__device__ __forceinline__ float sigmoidf_(float x) { return 1.f / (1.f + __expf(-x)); }

// order-preserving float <-> uint mapping for atomicMin-based segment-min
__device__ __forceinline__ unsigned fenc(float f) {
  unsigned u = __float_as_uint(f);
  return (u & 0x80000000u) ? ~u : (u | 0x80000000u);
}
__device__ __forceinline__ float fdec(unsigned k) {
  unsigned u = (k & 0x80000000u) ? (k ^ 0x80000000u) : ~k;
  return __uint_as_float(u);
}

// 16x128 = A(16x128, LDS) x W^T(128x128, W row-major in global)
// CDNA5 f32 WMMA 16x16x4: lane = (row = lane&15, hi = lane>>4)
// A frag: {A[row][4kk+2hi], A[row][4kk+2hi+1]} ; B frag: {W[16j+row][4kk+2hi], +1}
__device__ __forceinline__ void tile_gemm(const float* A, const float* __restrict__ W,
                                          v8f acc[8], int row, int hi) {
  const int coff = 2 * hi;
  for (int kk = 0; kk < 32; ++kk) {
    const int c0 = 4 * kk + coff;
    v2f a; a.x = A[row * LDSP + c0]; a.y = A[row * LDSP + c0 + 1];
#pragma unroll
    for (int j = 0; j < 8; ++j) {
      const float* wp = W + (16 * j + row) * D + c0;
      v2f b; b.x = wp[0]; b.y = wp[1];
      acc[j] = __builtin_amdgcn_wmma_f32_16x16x4_f32(false, a, false, b,
                                                     (short)0, acc[j], false, false);
    }
  }
}

// C/D layout: lane holds (m = v + 8*hi, n = 16*j + row)
__device__ __forceinline__ void tile_store(float* T, const v8f acc[8], int row, int hi) {
#pragma unroll
  for (int j = 0; j < 8; ++j)
#pragma unroll
    for (int v = 0; v < 8; ++v)
      T[(v + 8 * hi) * LDSP + 16 * j + row] = acc[j][v];
}

// ---------------- setup kernels ----------------
__global__ void k_fill(unsigned* __restrict__ p, size_t n, unsigned val) {
  size_t i = (size_t)blockIdx.x * blockDim.x + threadIdx.x;
  size_t stride = (size_t)gridDim.x * blockDim.x;
  for (; i < n; i += stride) p[i] = val;
}

// meta: [0..7]=hist  [8..16]=offs  [17..24]=cursor  [25..33]=groupOffs  [34]=totalGroups
__global__ void k_hist(const int* __restrict__ types, const int* __restrict__ tgt,
                       int* __restrict__ meta, int* __restrict__ cnt, int E, int N) {
  int e = blockIdx.x * blockDim.x + threadIdx.x;
  if (e >= E) return;
  int r = types[e];
  atomicAdd(&meta[r], 1);
  atomicAdd(&cnt[(size_t)r * N + tgt[e]], 1);
}

__global__ void k_scan(int* meta) {
  if (threadIdx.x != 0 || blockIdx.x != 0) return;
  int off = 0, goff = 0;
  meta[8] = 0; meta[25] = 0;
  for (int r = 0; r < NREL; ++r) {
    off  += meta[r];            meta[9 + r]  = off;
    goff += (meta[r] + 15) / 16; meta[26 + r] = goff;
  }
  meta[34] = goff;
}

__global__ void k_scatter(const int* __restrict__ types, int* __restrict__ meta,
                          int* __restrict__ perm, int E) {
  int e = blockIdx.x * blockDim.x + threadIdx.x;
  if (e >= E) return;
  int r = types[e];
  int pos = meta[8 + r] + atomicAdd(&meta[17 + r], 1);
  perm[pos] = e;
}

// ---------------- edge phase: per-relation WMMA transform + segment scatter ----------------
__global__ void __launch_bounds__(32)
k_edges(const float* __restrict__ h, const float* __restrict__ W_rel,
        const float* __restrict__ gate_W, const float* __restrict__ gate_b,
        const int* __restrict__ edge_index, const float* __restrict__ ew,
        const int* __restrict__ perm, const int* __restrict__ meta,
        unsigned* __restrict__ minU, float* __restrict__ sums, int N, int E) {
  __shared__ float Ht[16 * LDSP];
  __shared__ float Mt[16 * LDSP];
  __shared__ int srcA[16];
  __shared__ int tgtA[16];
  __shared__ float wA[16];

  int bid = blockIdx.x;
  if (bid >= meta[34]) return;           // beyond last group (uniform)
  int r = 0;
  while (r < 7 && bid >= meta[26 + r]) ++r;   // find relation bucket
  int lg   = bid - meta[25 + r];
  int base = meta[8 + r] + 16 * lg;
  int cg   = min(16, meta[r] - 16 * lg);

  const int lane = threadIdx.x, row = lane & 15, hi = lane >> 4;

  if (lane < 16) {
    int s = -1, t = -1; float w = 0.f;
    if (lane < cg) {
      int e = perm[base + lane];
      s = edge_index[e]; t = edge_index[E + e]; w = ew[e];
    }
    srcA[lane] = s; tgtA[lane] = t; wA[lane] = w;
  }
  __syncthreads();

  // gather 16 source rows (16x128 f32) into LDS, zero-fill padding rows
  for (int idx = lane; idx < 16 * 32; idx += 32) {
    int rr = idx >> 5, c4 = (idx & 31) * 4;
    float4 v = {0.f, 0.f, 0.f, 0.f};
    int s = srcA[rr];
    if (s >= 0) v = *(const float4*)(h + (size_t)s * D + c4);
    *(float4*)(Ht + rr * LDSP + c4) = v;
  }
  __syncthreads();

  const float* Wr = W_rel + (size_t)r * D * D;
  v8f acc[8] = {};
  tile_gemm(Ht, Wr, acc, row, hi);       // M = H_src * W_r^T

  if (r < 3) {                            // AND relations: gate GEMM + segment-min
    tile_store(Mt, acc, row, hi);
    __syncthreads();
    v8f g[8] = {};
    tile_gemm(Mt, gate_W, g, row, hi);   // G = M * gate_W^T
#pragma unroll
    for (int j = 0; j < 8; ++j) {
      int n = 16 * j + row;
      float bn = gate_b[n];
#pragma unroll
      for (int v = 0; v < 8; ++v) {
        int m = v + 8 * hi;
        int t = tgtA[m];
        if (t >= 0) {
          float mg  = Mt[m * LDSP + n];
          float val = mg * sigmoidf_(g[j][v] + bn) * wA[m];
          atomicMin(&minU[(((size_t)r * N + t) << 7) + n], fenc(val));
        }
      }
    }
  } else {                                // sum-based relations: segment-sum
    int si = r - 3;
#pragma unroll
    for (int j = 0; j < 8; ++j) {
      int n = 16 * j + row;
#pragma unroll
      for (int v = 0; v < 8; ++v) {
        int m = v + 8 * hi;
        int t = tgtA[m];
        if (t >= 0)
          atomicAdd(&sums[(((size_t)si * N + t) << 7) + n], acc[j][v]);
      }
    }
  }
}

// ---------------- node phase: self GEMM, IF attention, combine, ReLU+LN ----------------
__global__ void __launch_bounds__(32)
k_nodes(const float* __restrict__ h, const float* __restrict__ sitf,
        const float* __restrict__ W_rel, const float* __restrict__ W_self,
        const float* __restrict__ W_prem, const float* __restrict__ W_conc,
        const float* __restrict__ sit_W, const float* __restrict__ sit_b,
        const float* __restrict__ ln_g, const float* __restrict__ ln_b,
        const unsigned* __restrict__ minU, const float* __restrict__ sums,
        const int* __restrict__ cnt, float* __restrict__ out, int N) {
  __shared__ float Ht[16 * LDSP];
  __shared__ float Out[16 * LDSP];
  __shared__ float Sa[16 * LDSP];
  __shared__ float Sb[16 * LDSP];
  __shared__ float Sc[16 * LDSP];
  __shared__ float attnL[16];

  const int lane = threadIdx.x, row = lane & 15, hi = lane >> 4;
  const int nb = blockIdx.x * 16;

  for (int idx = lane; idx < 16 * 32; idx += 32) {
    int m = idx >> 5, c4 = (idx & 31) * 4;
    int node = nb + m;
    float4 v = {0.f, 0.f, 0.f, 0.f};
    if (node < N) v = *(const float4*)(h + (size_t)node * D + c4);
    *(float4*)(Ht + m * LDSP + c4) = v;
  }
  __syncthreads();

  {
    v8f acc[8] = {};
    tile_gemm(Ht, W_self, acc, row, hi);  // self-loop
    tile_store(Out, acc, row, hi);
  }

  // IF relations r=3,4: premise/attention gating
  for (int rr = 3; rr <= 4; ++rr) {
    const float* Wr = W_rel + (size_t)rr * D * D;
    v8f t[8] = {};
    tile_gemm(Ht, Wr, t, row, hi);        // tgt_h
    tile_store(Sa, t, row, hi);
    for (int idx = lane; idx < 2048; idx += 32) {  // premise = sum / max(cnt,1)
      int m = idx >> 7, n = idx & 127, node = nb + m;
      float p = 0.f;
      if (node < N) {
        float c = (float)cnt[(size_t)rr * N + node];
        p = sums[(((size_t)(rr - 3) * N + node) << 7) + n] / fmaxf(c, 1.f);
      }
      Sb[m * LDSP + n] = p;
    }
    __syncthreads();
    v8f pp[8] = {};
    tile_gemm(Sb, W_prem, pp, row, hi);   // premise * W_premise^T
    tile_store(Sc, pp, row, hi);
    if (lane < 16) attnL[lane] = 0.f;
    __syncthreads();
    v8f qq[8] = {};
    tile_gemm(Sa, W_conc, qq, row, hi);   // tgt_h * W_conclusion^T
    float part[8];
#pragma unroll
    for (int v = 0; v < 8; ++v) part[v] = 0.f;
#pragma unroll
    for (int j = 0; j < 8; ++j)
#pragma unroll
      for (int v = 0; v < 8; ++v)
        part[v] += Sc[(v + 8 * hi) * LDSP + 16 * j + row] * qq[j][v];
#pragma unroll
    for (int v = 0; v < 8; ++v) atomicAdd(&attnL[v + 8 * hi], part[v]);
    __syncthreads();
    const float inv_scale = 0.088388347648f;  // 1/sqrt(128)
    for (int idx = lane; idx < 2048; idx += 32) {
      int m = idx >> 7, n = idx & 127, node = nb + m;
      if (node < N && cnt[(size_t)rr * N + node] > 0) {
        float g = sigmoidf_(attnL[m] * inv_scale);
        Out[m * LDSP + n] += Sa[m * LDSP + n] * g + Sb[m * LDSP + n] * (1.f - g);
      }
    }
    __syncthreads();
  }

  // element-wise messages: AND mins (r=0..2), IST (r=5), means (r=6,7)
  for (int idx = lane; idx < 2048; idx += 32) {
    int m = idx >> 7, n = idx & 127, node = nb + m;
    if (node >= N) continue;
    float o = Out[m * LDSP + n];
#pragma unroll
    for (int r = 0; r < 3; ++r)
      if (cnt[(size_t)r * N + node] > 0)
        o += fdec(minU[(((size_t)r * N + node) << 7) + n]);
    int c5 = cnt[(size_t)5 * N + node];
    if (c5 > 0) {
      float mean = sums[(((size_t)2 * N + node) << 7) + n] / (float)c5;
      float dv = sit_b[n];
#pragma unroll
      for (int k = 0; k < 16; ++k) dv += sitf[(size_t)node * 16 + k] * sit_W[n * 16 + k];
      o += mean * sigmoidf_(dv);
    }
#pragma unroll
    for (int r = 6; r <= 7; ++r) {
      int c = cnt[(size_t)r * N + node];
      if (c > 0) o += sums[(((size_t)(r - 3) * N + node) << 7) + n] / (float)c;
    }
    Out[m * LDSP + n] = o;
  }
  __syncthreads();

  // ReLU + LayerNorm (wave32 shuffle reduction, 4 elems/lane per row)
  for (int m = 0; m < 16; ++m) {
    int node = nb + m;
    if (node >= N) break;
    float x[4], s = 0.f, ss = 0.f;
#pragma unroll
    for (int k = 0; k < 4; ++k) {
      float xv = fmaxf(Out[m * LDSP + lane + 32 * k], 0.f);
      x[k] = xv; s += xv; ss += xv * xv;
    }
#pragma unroll
    for (int d_ = 16; d_ >= 1; d_ >>= 1) {
      s  += __shfl_xor(s, d_);
      ss += __shfl_xor(ss, d_);
    }
    float mu   = s * (1.f / 128.f);
    float var  = ss * (1.f / 128.f) - mu * mu;
    float rstd = rsqrtf(var + 1e-5f);
#pragma unroll
    for (int k = 0; k < 4; ++k) {
      int n = lane + 32 * k;
      out[(size_t)node * D + n] = (x[k] - mu) * rstd * ln_g[n] + ln_b[n];
    }
  }
}

// ---------------- host ----------------
extern "C" void kernel_launch(void* const* d_in, const int* in_sizes, int n_in,
                              void* d_out, int out_size, void* d_ws, size_t ws_size,
                              hipStream_t stream) {
  const float* h      = (const float*)d_in[0];
  const int*   ei     = (const int*)d_in[1];   // (2,E)
  const int*   et     = (const int*)d_in[2];   // (E,)
  const float* ew     = (const float*)d_in[3];
  const float* sitf   = (const float*)d_in[4];
  const float* W_rel  = (const float*)d_in[5];
  const float* W_self = (const float*)d_in[6];
  const float* gate_W = (const float*)d_in[7];
  const float* gate_b = (const float*)d_in[8];
  const float* W_prem = (const float*)d_in[9];
  const float* W_conc = (const float*)d_in[10];
  const float* sit_W  = (const float*)d_in[11];
  const float* sit_b  = (const float*)d_in[12];
  const float* ln_g   = (const float*)d_in[13];
  const float* ln_b   = (const float*)d_in[14];
  const int N = in_sizes[0] / D;
  const int E = in_sizes[2];
  float* out = (float*)d_out;
  (void)n_in; (void)out_size; (void)ws_size;

  char* ws = (char*)d_ws;
  size_t off = 0;
  auto take = [&](size_t bytes) -> char* {
    char* p = ws + off;
    off += (bytes + 255) & ~(size_t)255;
    return p;
  };
  int*      meta = (int*)take(64 * sizeof(int));
  int*      cnt  = (int*)take((size_t)NREL * N * sizeof(int));
  int*      perm = (int*)take((size_t)E * sizeof(int));
  unsigned* minU = (unsigned*)take((size_t)3 * N * D * sizeof(unsigned));
  float*    sums = (float*)take((size_t)5 * N * D * sizeof(float));

  hipMemsetAsync(meta, 0, 64 * sizeof(int), stream);
  hipMemsetAsync(cnt, 0, (size_t)NREL * N * sizeof(int), stream);
  hipMemsetAsync(sums, 0, (size_t)5 * N * D * sizeof(float), stream);
  k_fill<<<4096, 256, 0, stream>>>(minU, (size_t)3 * N * D, 0xFF800000u);  // fenc(+inf)

  k_hist<<<(E + 255) / 256, 256, 0, stream>>>(et, ei + E, meta, cnt, E, N);
  k_scan<<<1, 1, 0, stream>>>(meta);
  k_scatter<<<(E + 255) / 256, 256, 0, stream>>>(et, meta, perm, E);
  k_edges<<<E / 16 + NREL, 32, 0, stream>>>(h, W_rel, gate_W, gate_b, ei, ew,
                                            perm, meta, minU, sums, N, E);
  k_nodes<<<(N + 15) / 16, 32, 0, stream>>>(h, sitf, W_rel, W_self, W_prem, W_conc,
                                            sit_W, sit_b, ln_g, ln_b,
                                            minU, sums, cnt, out, N);
}